// Mapping_2_37177236914282
// MI455X (gfx1250) — compile-verified
//
#include <hip/hip_runtime.h>

// MI455X / gfx1250, wave32. Bandwidth-bound masked-vote reduction:
// ~587MB read / launch -> roofline ~25us @ 23.3 TB/s.
// Cross-lane f32 reductions are done on the matrix pipe with
// V_WMMA_F32_16X16X4_F32 (A = per-lane data, B = ones, C = 0).

typedef __attribute__((ext_vector_type(2))) float v2f;
typedef __attribute__((ext_vector_type(8))) float v8f;

// Full 32-lane f32 sum using one WMMA + 7 VALU adds + 1 shuffle.
// A layout (16x4 f32): VGPR0 lanes0-15 -> K=0 (=v[m]), lanes16-31 -> K=2 (=v[m+16]);
// VGPR1 -> K=1,3 (zeroed). B = all ones => D[m][n] = v[m] + v[m+16].
// Sum of 8 D VGPRs gives half-sums in lane halves; shfl_xor(16) completes it.
__device__ __forceinline__ float wave_sum32(float v) {
  v2f a;    a[0] = v;    a[1] = 0.0f;
  v2f ones; ones[0] = 1.0f; ones[1] = 1.0f;
  v8f c = {};
  v8f d = __builtin_amdgcn_wmma_f32_16x16x4_f32(
      /*neg_a=*/false, a, /*neg_b=*/false, ones,
      /*c_mod=*/(short)0, c, /*reuse_a=*/false, /*reuse_b=*/false);
  float s = ((d[0] + d[1]) + (d[2] + d[3])) + ((d[4] + d[5]) + (d[6] + d[7]));
  s += __shfl_xor(s, 16, 32);
  return s;
}

__global__ __launch_bounds__(640)
void hand_kp_kernel(const float* __restrict__ fvec, const float* __restrict__ fdis,
                    const float* __restrict__ bvec, const float* __restrict__ bdis,
                    const float* __restrict__ mask, float* __restrict__ out) {
  const int b    = blockIdx.x;
  const int w    = threadIdx.x >> 5;   // wave == channel 0..19
  const int lane = threadIdx.x & 31;

  const size_t p = ((size_t)b * 20 + w) * 4096;  // spatial base of this (b,c)
  const float4* mk4 = (const float4*)(mask + p);
  const float4* fd4 = (const float4*)(fdis + p);
  const float4* bd4 = (const float4*)(bdis + p);
  const float4* fv4 = (const float4*)(fvec + p * 2);
  const float4* bv4 = (const float4*)(bvec + p * 2);

  float afx = 0.f, afy = 0.f, abx = 0.f, aby = 0.f, alx = 0.f, aly = 0.f, am = 0.f;

#pragma unroll 4
  for (int i = 0; i < 32; ++i) {
    const int s0 = i * 128 + lane * 4;      // 4 consecutive spatial positions
    const int q  = s0 >> 2;
    const float4 m4 = mk4[q];
    const float4 f4 = fd4[q];
    const float4 b4 = bd4[q];
    const float4 fa = fv4[s0 >> 1];         // (x,y) for s0, s0+1
    const float4 fb = fv4[(s0 >> 1) + 1];   // (x,y) for s0+2, s0+3
    const float4 ga = bv4[s0 >> 1];
    const float4 gb = bv4[(s0 >> 1) + 1];

    const float fw0 = m4.x * f4.x, fw1 = m4.y * f4.y, fw2 = m4.z * f4.z, fw3 = m4.w * f4.w;
    const float bw0 = m4.x * b4.x, bw1 = m4.y * b4.y, bw2 = m4.z * b4.z, bw3 = m4.w * b4.w;

    afx += fa.x * fw0 + fa.z * fw1 + fb.x * fw2 + fb.z * fw3;
    afy += fa.y * fw0 + fa.w * fw1 + fb.y * fw2 + fb.w * fw3;
    abx += ga.x * bw0 + ga.z * bw1 + gb.x * bw2 + gb.z * bw3;
    aby += ga.y * bw0 + ga.w * bw1 + gb.y * bw2 + gb.w * bw3;

    const float msum = (m4.x + m4.y) + (m4.z + m4.w);
    am  += msum;
    alx += (float)(s0 >> 6) * msum;                                // x = s>>6, same for all 4
    aly += (float)(s0 & 63) * msum + (m4.y + 2.0f * m4.z + 3.0f * m4.w);  // y = base+k
  }

  // Wave-level (EXEC all-ones here) f32 reductions on the matrix pipe.
  afx = wave_sum32(afx);  afy = wave_sum32(afy);
  abx = wave_sum32(abx);  aby = wave_sum32(aby);
  alx = wave_sum32(alx);  aly = wave_sum32(aly);
  am  = wave_sum32(am);

  __shared__ float ssum[20][8];
  if (lane == 0) {
    ssum[w][0] = afx; ssum[w][1] = afy; ssum[w][2] = abx; ssum[w][3] = aby;
    ssum[w][4] = alx; ssum[w][5] = aly; ssum[w][6] = am;
  }
  __syncthreads();

  __shared__ float2 sfront[20], sback[20];
  const int tid = threadIdx.x;
  if (tid < 20) {
    const float* qq = ssum[tid];
    const float den = qq[6] + 1e-6f;
    sfront[tid] = make_float2((64.0f * qq[0] + qq[4]) / den, (64.0f * qq[1] + qq[5]) / den);
    sback[tid]  = make_float2((64.0f * qq[2] + qq[4]) / den, (64.0f * qq[3] + qq[5]) / den);
  }
  __syncthreads();

  if (tid < 21) {
    float2* o = (float2*)out + (size_t)b * 21;
    if (tid == 20) {
      // wrist: mean of back channels 0,4,8,12,16, then *4  => *0.8
      float sx = 0.f, sy = 0.f;
      for (int i = 0; i < 20; i += 4) { sx += sback[i].x; sy += sback[i].y; }
      o[0] = make_float2(sx * 0.8f, sy * 0.8f);
    } else {
      const int cc = tid, j = cc & 3;
      const int t = (cc & ~3) + 4 - j;           // target keypoint 1..20 (bijective)
      float2 v;
      if (j == 3) {
        v = sfront[cc];
      } else {
        v.x = 0.5f * (sfront[cc].x + sback[cc + 1].x);
        v.y = 0.5f * (sfront[cc].y + sback[cc + 1].y);
      }
      o[t] = make_float2(v.x * 4.0f, v.y * 4.0f);
    }
  }
}

extern "C" void kernel_launch(void* const* d_in, const int* in_sizes, int n_in,
                              void* d_out, int out_size, void* d_ws, size_t ws_size,
                              hipStream_t stream) {
  (void)n_in; (void)d_ws; (void)ws_size; (void)out_size;
  const float* fvec = (const float*)d_in[0];  // [B,20,64,64,2]
  const float* fdis = (const float*)d_in[1];  // [B,20,64,64,1]
  const float* bvec = (const float*)d_in[2];  // [B,20,64,64,2]
  const float* bdis = (const float*)d_in[3];  // [B,20,64,64,1]
  const float* mask = (const float*)d_in[4];  // [B,20,64,64,1]
  float* out = (float*)d_out;                 // [B,21,2]

  const int B = in_sizes[0] / (20 * 64 * 64 * 2);
  hand_kp_kernel<<<dim3(B), dim3(640), 0, stream>>>(fvec, fdis, bvec, bdis, mask, out);
}